// PatchContextualLoss_89421219103076
// MI455X (gfx1250) — compile-verified
//
#include <hip/hip_runtime.h>
#include <math.h>

typedef __attribute__((ext_vector_type(2))) float v2f;
typedef __attribute__((ext_vector_type(8))) float v8f;

#define IMG 192
#define G96 96
#define HH  24
#define P   576
#define C   25
#define CP  32
#define NB  16

// workspace layout (float offsets)
#define WS_MEANT 0
#define WS_FI    32
#define WS_FT    (32 + NB*P*CP)
#define WS_CXSP  (32 + 2*NB*P*CP)
#define WS_KMAX  (32 + 2*NB*P*CP + P*P)

// ---------------- kernel 0: zero column-max accumulator ----------------
__global__ void k_zero(unsigned* __restrict__ kmax) {
  int i = blockIdx.x * blockDim.x + threadIdx.x;
  if (i < NB * P) kmax[i] = 0u;
}

// ---------------- kernel 1: per-channel mean of gt patches ----------------
__global__ __launch_bounds__(256) void k_meant(const float* __restrict__ gt,
                                               float* __restrict__ meanT) {
  int c = blockIdx.x;           // 0..24
  int kh = c / 5, kw = c % 5;
  float s = 0.f;
  for (int idx = threadIdx.x; idx < G96 * G96; idx += 256) {
    int y = idx / G96, x = idx % G96;
    int r = 2 * y - 2 + kh, q = 2 * x - 2 + kw;
    if (r >= 0 && r < IMG && q >= 0 && q < IMG) s += gt[r * IMG + q];
  }
  __shared__ float red[256];
  red[threadIdx.x] = s;
  __syncthreads();
  for (int off = 128; off > 0; off >>= 1) {
    if (threadIdx.x < off) red[threadIdx.x] += red[threadIdx.x + off];
    __syncthreads();
  }
  if (threadIdx.x == 0) meanT[c] = red[0] / 9216.0f;
}

// ---------------- kernel 2: centered + L2-normalized patch features ----------------
__global__ __launch_bounds__(256) void k_feat(const float* __restrict__ im,
                                              const float* __restrict__ gt,
                                              const float* __restrict__ meanT,
                                              float* __restrict__ Fi,
                                              float* __restrict__ Ft) {
  int tid = blockIdx.x * 256 + threadIdx.x;
  if (tid >= 2 * NB * P) return;
  int img = tid / (NB * P);
  int rem = tid % (NB * P);
  int n = rem / P, p = rem % P;
  int h = p / HH, w = p % HH;
  int n1 = n & 3, q2 = n >> 2;      // double crop_quarters: n = q2*4 + q1
  int y = 48 * ((n1 >> 1) & 1) + 24 * ((q2 >> 1) & 1) + h;
  int x = 48 * (n1 & 1)        + 24 * (q2 & 1)        + w;
  const float* src = img ? gt : im;
  float f[C];
  float nrm2 = 0.f;
#pragma unroll
  for (int c = 0; c < C; c++) {
    int kh = c / 5, kw = c % 5;
    int r = 2 * y - 2 + kh, q = 2 * x - 2 + kw;
    float v = (r >= 0 && r < IMG && q >= 0 && q < IMG) ? src[r * IMG + q] : 0.f;
    v -= meanT[c];
    f[c] = v;
    nrm2 += v * v;
  }
  float nrm = sqrtf(nrm2);
  float* dst = (img ? Ft : Fi) + (size_t)(n * P + p) * CP;
#pragma unroll
  for (int c = 0; c < C; c++) dst[c] = f[c] / nrm;
#pragma unroll
  for (int c = C; c < CP; c++) dst[c] = 0.f;
}

// ---------------- kernel 3: spatial CX matrix via V_WMMA_F32_16X16X4_F32 ----------------
// raw_sp[i,j] = |g_i - g_j|^2 = A x B with K=4:
//   A[i,:] = [r_i, c_i, s_i, 1], B[:,j] = [-2 r_j, -2 c_j, 1, s_j]
__global__ __launch_bounds__(32) void k_cxsp(float* __restrict__ cxsp) {
  __shared__ float sRaw[16][P];      // 36 KB
  __shared__ float sMin[16][2];
  __shared__ float sSum[16][2];
  int lane = threadIdx.x;
  int i0 = blockIdx.x * 16;
  int m = lane & 15;
  int half = lane >> 4;
  int i = i0 + m;
  float ri = (float)(i / HH) / 25.0f;
  float ci = (float)(i % HH) / 25.0f;
  float si = ri * ri + ci * ci;
  // A 16x4 f32: lanes 0-15 -> {K0,K1}, lanes 16-31 -> {K2,K3}
  v2f a;
  a.x = (lane < 16) ? ri : si;
  a.y = (lane < 16) ? ci : 1.0f;

  for (int j0 = 0; j0 < P; j0 += 16) {
    int j = j0 + m;
    float rj = (float)(j / HH) / 25.0f;
    float cj = (float)(j % HH) / 25.0f;
    float sj = rj * rj + cj * cj;
    // B 4x16 f32: VGPR0 -> rows K0 (lanes 0-15) / K2 (lanes 16-31); VGPR1 -> K1 / K3
    v2f b;
    b.x = (lane < 16) ? (-2.0f * rj) : 1.0f;
    b.y = (lane < 16) ? (-2.0f * cj) : sj;
    v8f cz = {};
    v8f d = __builtin_amdgcn_wmma_f32_16x16x4_f32(
        /*neg_a=*/false, a, /*neg_b=*/false, b,
        /*c_mod=*/(short)0, cz, /*reuse_a=*/false, /*reuse_b=*/false);
    int rbase = half * 8;          // D: VGPR v -> row v (lanes0-15) / v+8 (lanes16-31)
#pragma unroll
    for (int v = 0; v < 8; v++) sRaw[rbase + v][j0 + m] = fmaxf(d[v], 0.0f);
  }
  __syncthreads();

  // row-wise: dmin -> w = exp((1-d~)*2) -> rowsum -> normalize. Two lanes per row.
  int jb = half * 288;
  float mn = 3.4e38f;
  for (int j = jb; j < jb + 288; j++) mn = fminf(mn, sRaw[m][j]);
  sMin[m][half] = mn;
  __syncthreads();
  float denom = fminf(sMin[m][0], sMin[m][1]) + 1e-5f;
  float s = 0.f;
  for (int j = jb; j < jb + 288; j++) {
    float w = expf((1.0f - sRaw[m][j] / denom) * 2.0f);
    sRaw[m][j] = w;
    s += w;
  }
  sSum[m][half] = s;
  __syncthreads();
  float rowsum = sSum[m][0] + sSum[m][1];
  for (int j = jb; j < jb + 288; j++)
    cxsp[(size_t)(i0 + m) * P + j] = sRaw[m][j] / rowsum;
}

// ---------------- kernel 4: fused L1 distance + CX + column-max ----------------
__global__ __launch_bounds__(256) void k_main(const float* __restrict__ Fi,
                                              const float* __restrict__ Ft,
                                              const float* __restrict__ cxsp,
                                              unsigned* __restrict__ kmax) {
  __shared__ float sFi[16][CP];      // 2 KB
  __shared__ float sFt[64][CP];      // 8 KB (async-filled)
  __shared__ float sRaw[16][P];      // 36 KB
  __shared__ float sRed[16][16];
  __shared__ float sDmin[16];
  __shared__ float sRsum[16];

  int n = blockIdx.y;
  int i0 = blockIdx.x * 16;
  int t = threadIdx.x;

  for (int idx = t; idx < 16 * CP; idx += 256) {
    int row = idx >> 5, c = idx & 31;
    sFi[row][c] = Fi[(size_t)(n * P + i0 + row) * CP + c];
  }
  __syncthreads();

  const float* FtB = Ft + (size_t)(n * P) * CP;
  int ii = t & 15;
  int jb = t >> 4;
  for (int j0 = 0; j0 < P; j0 += 64) {
    // async stage the 64x25 Tv tile into LDS (ASYNCcnt path)
#pragma unroll
    for (int k = 0; k < 2; k++) {
      int q = t + 256 * k;
      int row = q >> 3, f4 = (q & 7) * 4;
      unsigned lds = (unsigned)(size_t)&sFt[row][f4];
      const float* g = FtB + (size_t)(j0 + row) * CP + f4;
      asm volatile("global_load_async_to_lds_b128 %0, %1, off"
                   :: "v"(lds), "v"(g) : "memory");
    }
    asm volatile("s_wait_asynccnt 0x0" ::: "memory");
    __syncthreads();
#pragma unroll
    for (int k = 0; k < 4; k++) {
      int jj = jb * 4 + k;
      float acc = 0.f;
#pragma unroll
      for (int c = 0; c < C; c++) acc += fabsf(sFi[ii][c] - sFt[jj][c]);
      sRaw[ii][j0 + jj] = acc;
    }
    __syncthreads();
  }

  // per-row dmin
  int row = t & 15, seg = t >> 4;
  float mn = 3.4e38f;
  for (int k = 0; k < 36; k++) mn = fminf(mn, sRaw[row][seg * 36 + k]);
  sRed[row][seg] = mn;
  __syncthreads();
  if (t < 16) {
    float m2 = sRed[t][0];
    for (int k = 1; k < 16; k++) m2 = fminf(m2, sRed[t][k]);
    sDmin[t] = m2 + 1e-5f;
  }
  __syncthreads();

  // w = exp((1 - d~)*2), rowsum
  float denom = sDmin[row];
  float s = 0.f;
  for (int k = 0; k < 36; k++) {
    int j = seg * 36 + k;
    float w = expf((1.0f - sRaw[row][j] / denom) * 2.0f);
    sRaw[row][j] = w;
    s += w;
  }
  sRed[row][seg] = s;
  __syncthreads();
  if (t < 16) {
    float t2 = 0.f;
    for (int k = 0; k < 16; k++) t2 += sRed[t][k];
    sRsum[t] = t2;
  }
  __syncthreads();

  // cx = 0.9*cx_feat + 0.1*cx_sp ; column-max over this block's 16 rows, merge via atomic
  for (int j = t; j < P; j += 256) {
    float cm = 0.f;
#pragma unroll
    for (int r2 = 0; r2 < 16; r2++) {
      float cx = 0.9f * sRaw[r2][j] / sRsum[r2]
               + 0.1f * cxsp[(size_t)(i0 + r2) * P + j];
      cm = fmaxf(cm, cx);
    }
    atomicMax(&kmax[n * P + j], __float_as_uint(cm));  // non-negative -> bit-compare safe
  }
}

// ---------------- kernel 5: cs = mean_j kmax ; loss = mean_n -log(cs) ----------------
__global__ __launch_bounds__(256) void k_final(const unsigned* __restrict__ kmax,
                                               float* __restrict__ out) {
  __shared__ float red[256];
  __shared__ float acc;
  if (threadIdx.x == 0) acc = 0.f;
  __syncthreads();
  for (int n = 0; n < NB; n++) {
    float s = 0.f;
    for (int j = threadIdx.x; j < P; j += 256) s += __uint_as_float(kmax[n * P + j]);
    red[threadIdx.x] = s;
    __syncthreads();
    for (int off = 128; off > 0; off >>= 1) {
      if (threadIdx.x < off) red[threadIdx.x] += red[threadIdx.x + off];
      __syncthreads();
    }
    if (threadIdx.x == 0) acc += -logf(red[0] / 576.0f);
    __syncthreads();
  }
  if (threadIdx.x == 0) out[0] = acc / 16.0f;
}

extern "C" void kernel_launch(void* const* d_in, const int* in_sizes, int n_in,
                              void* d_out, int out_size, void* d_ws, size_t ws_size,
                              hipStream_t stream) {
  (void)in_sizes; (void)n_in; (void)out_size; (void)ws_size;
  const float* im = (const float*)d_in[0];
  const float* gt = (const float*)d_in[1];
  float* ws = (float*)d_ws;
  float* meanT = ws + WS_MEANT;
  float* Fi    = ws + WS_FI;
  float* Ft    = ws + WS_FT;
  float* cxsp  = ws + WS_CXSP;
  unsigned* kmax = (unsigned*)(ws + WS_KMAX);
  float* out = (float*)d_out;

  hipLaunchKernelGGL(k_zero,  dim3(36), dim3(256), 0, stream, kmax);
  hipLaunchKernelGGL(k_meant, dim3(25), dim3(256), 0, stream, gt, meanT);
  hipLaunchKernelGGL(k_feat,  dim3((2 * NB * P + 255) / 256), dim3(256), 0, stream,
                     im, gt, meanT, Fi, Ft);
  hipLaunchKernelGGL(k_cxsp,  dim3(36), dim3(32), 0, stream, cxsp);
  hipLaunchKernelGGL(k_main,  dim3(36, 16), dim3(256), 0, stream, Fi, Ft, cxsp, kmax);
  hipLaunchKernelGGL(k_final, dim3(1), dim3(256), 0, stream, kmax, out);
}